// GAT_gate_67937792688477
// MI455X (gfx1250) — compile-verified
//
#include <hip/hip_runtime.h>
#include <hip/hip_bf16.h>
#include <math.h>

#define BB 16
#define NN 1024
#define DD 128
#define NHOP 4
#define KC 64              // k-chunk rows staged in LDS per block

typedef float v2f __attribute__((ext_vector_type(2)));
typedef float v4f __attribute__((ext_vector_type(4)));
typedef float v8f __attribute__((ext_vector_type(8)));

// D = A(16x4) * B(4x16) + C  on the f32 matrix pipe
__device__ __forceinline__ v8f wmma_f32(v2f a, v2f b, v8f c) {
    return __builtin_amdgcn_wmma_f32_16x16x4_f32(
        false, a, false, b, (short)0, c, false, false);
}

// ---------------------------------------------------------------------------
// Kernel 1: h = x @ W_w^T + W_b ; hA = h @ A       (one wave per 16-row tile)
// ---------------------------------------------------------------------------
__global__ void __launch_bounds__(32)
gat_proj_kernel(const float* __restrict__ x, const float* __restrict__ Ww,
                const float* __restrict__ Wb, const float* __restrict__ Amat,
                float* __restrict__ h, float* __restrict__ hA) {
    const int tile = blockIdx.x;           // B * N/16 tiles
    const int b  = tile >> 6;              // / 64
    const int n0 = (tile & 63) << 4;
    const int lane    = threadIdx.x;
    const int laneRow = lane & 15;
    const int laneHi  = lane >> 4;

    // Preload all A-fragments of the x row-tile (16x128): 32 k-steps
    v2f ax[32];
#pragma unroll
    for (int kk = 0; kk < 32; ++kk) {
        const int k = kk * 4 + laneHi * 2;
        ax[kk] = *(const v2f*)&x[((size_t)(b * NN + n0 + laneRow)) * DD + k];
    }

    // h = x @ Ww^T + bias   (B[l, c] = Ww[c, l] : contiguous in l)
#pragma unroll 1
    for (int ct = 0; ct < 8; ++ct) {
        const int c0 = ct * 16;
        v8f acc = {};
#pragma unroll
        for (int kk = 0; kk < 32; ++kk) {
            const int k = kk * 4 + laneHi * 2;
            v2f bf = *(const v2f*)&Ww[(size_t)(c0 + laneRow) * DD + k];
            acc = wmma_f32(ax[kk], bf, acc);
        }
        const float bias = Wb[c0 + laneRow];
#pragma unroll
        for (int v = 0; v < 8; ++v) {
            const int r = n0 + v + laneHi * 8;
            h[((size_t)(b * NN + r)) * DD + c0 + laneRow] = acc[v] + bias;
        }
    }

    // make our own stores visible before re-reading the tile as A-fragments
    asm volatile("s_wait_storecnt 0" ::: "memory");

    v2f ah[32];
#pragma unroll
    for (int kk = 0; kk < 32; ++kk) {
        const int k = kk * 4 + laneHi * 2;
        ah[kk] = *(const v2f*)&h[((size_t)(b * NN + n0 + laneRow)) * DD + k];
    }

    // hA = h @ A   (B[l, c] = A[l, c] : stride D between the two k elements)
#pragma unroll 1
    for (int ct = 0; ct < 8; ++ct) {
        const int c0 = ct * 16;
        v8f acc = {};
#pragma unroll
        for (int kk = 0; kk < 32; ++kk) {
            const int k = kk * 4 + laneHi * 2;
            v2f bf;
            bf.x = Amat[(size_t)(k    ) * DD + c0 + laneRow];
            bf.y = Amat[(size_t)(k + 1) * DD + c0 + laneRow];
            acc = wmma_f32(ah[kk], bf, acc);
        }
#pragma unroll
        for (int v = 0; v < 8; ++v) {
            const int r = n0 + v + laneHi * 8;
            hA[((size_t)(b * NN + r)) * DD + c0 + laneRow] = acc[v];
        }
    }
}

// ---------------------------------------------------------------------------
// Kernel 2: att_raw[j,k] = adj>0 ? (hA_j.h_k + hA_k.h_j) : 0
//           one wave per 16x16 output tile; e + e^T fused into one accumulator
// ---------------------------------------------------------------------------
__global__ void __launch_bounds__(32)
gat_scores_kernel(const float* __restrict__ h, const float* __restrict__ hA,
                  const float* __restrict__ adj, float* __restrict__ att) {
    const int id  = blockIdx.x;            // B * 64 * 64
    const int b   = id >> 12;
    const int jt  = (id >> 6) & 63;
    const int kt  = id & 63;
    const int j0 = jt << 4, k0 = kt << 4;
    const int lane    = threadIdx.x;
    const int laneRow = lane & 15;
    const int laneHi  = lane >> 4;

    const size_t rowJ = ((size_t)(b * NN + j0 + laneRow)) * DD;
    const size_t rowK = ((size_t)(b * NN + k0 + laneRow)) * DD;

    v8f acc = {};
#pragma unroll
    for (int kk = 0; kk < 32; ++kk) {
        const int k = kk * 4 + laneHi * 2;
        // e      : A = hA rows(j-tile), B = h^T  (B[l,c] = h[k0+c][l])
        v2f a1 = *(const v2f*)&hA[rowJ + k];
        v2f b1 = *(const v2f*)&h [rowK + k];
        acc = wmma_f32(a1, b1, acc);
        // e^T    : A = h rows(j-tile),  B = hA^T
        v2f a2 = *(const v2f*)&h [rowJ + k];
        v2f b2 = *(const v2f*)&hA[rowK + k];
        acc = wmma_f32(a2, b2, acc);
    }

#pragma unroll
    for (int v = 0; v < 8; ++v) {
        const int r = j0 + v + laneHi * 8;
        const int c = k0 + laneRow;
        const size_t idx = ((size_t)(b * NN + r)) * NN + c;
        const float av = adj[idx];
        att[idx] = (av > 0.0f) ? acc[v] : 0.0f;
    }
}

// ---------------------------------------------------------------------------
// Kernel 3: softmax over axis=1 (over j, per column k), then att *= adj
//           thread-per-column => coalesced over k on every j-row sweep
// ---------------------------------------------------------------------------
__global__ void __launch_bounds__(256)
gat_softmax_kernel(const float* __restrict__ adj, float* __restrict__ att) {
    const int b = blockIdx.x >> 2;                       // B * (N/256) blocks
    const int c = ((blockIdx.x & 3) << 8) + threadIdx.x; // column in [0,N)
    const size_t base = (size_t)b * NN * NN + c;

    float m = -INFINITY;
    for (int j = 0; j < NN; ++j)
        m = fmaxf(m, att[base + (size_t)j * NN]);

    float s = 0.0f;
    for (int j = 0; j < NN; ++j)
        s += __expf(att[base + (size_t)j * NN] - m);
    const float inv = 1.0f / s;

    for (int j = 0; j < NN; ++j) {
        const size_t idx = base + (size_t)j * NN;
        att[idx] = __expf(att[idx] - m) * inv * adj[idx];
    }
}

// ---------------------------------------------------------------------------
// Kernel 4 (x4 hops): az = relu(att @ z); coeff = sigmoid([x,az].gate_w + b);
//                     z' = coeff*x + (1-coeff)*az
// 8 waves / block, 128 rows / block; z staged chunk-wise in LDS (32KB) and
// shared by all 8 waves; att A-fragments preloaded per chunk and reused
// across all 8 column tiles.
// ---------------------------------------------------------------------------
__global__ void __launch_bounds__(256)
gat_hop_kernel(const float* __restrict__ att, const float* __restrict__ x,
               const float* __restrict__ zin, const float* __restrict__ gw,
               const float* __restrict__ gb, float* __restrict__ zout) {
    __shared__ float lz[KC * DD];                 // 64 x 128 f32 = 32KB

    const int b  = blockIdx.x >> 3;               // B * (N/128) blocks
    const int r0 = (blockIdx.x & 7) << 7;         // first row of block
    const int wave = threadIdx.x >> 5;
    const int lane = threadIdx.x & 31;
    const int laneRow = lane & 15;
    const int laneHi  = lane >> 4;
    const int i0 = r0 + wave * 16;                // this wave's 16 rows

    const size_t attRow = ((size_t)(b * NN + i0 + laneRow)) * NN;
    const size_t zBase  = (size_t)b * NN * DD;

    v8f az[8];
#pragma unroll
    for (int ct = 0; ct < 8; ++ct) az[ct] = (v8f){};

    for (int k0 = 0; k0 < NN; k0 += KC) {
        // ---- stage z[k0 .. k0+KC) x [0,D) into LDS (coalesced b128) ----
        {
            const v4f* zsrc = (const v4f*)&zin[zBase + (size_t)k0 * DD];
            v4f* ldst = (v4f*)lz;
#pragma unroll
            for (int i = 0; i < (KC * DD / 4) / 256; ++i)
                ldst[threadIdx.x + i * 256] = zsrc[threadIdx.x + i * 256];
        }
        __syncthreads();

        // ---- preload att A-fragments for this chunk (reused by all 8 ct) --
        v2f af[KC / 4];
#pragma unroll
        for (int kk = 0; kk < KC / 4; ++kk) {
            const int k = k0 + kk * 4 + laneHi * 2;
            af[kk] = *(const v2f*)&att[attRow + k];
        }

        // ---- accumulate all 8 column tiles from LDS ----
#pragma unroll 1
        for (int ct = 0; ct < 8; ++ct) {
            const int c0 = ct * 16;
            v8f acc = az[ct];
#pragma unroll
            for (int kk = 0; kk < KC / 4; ++kk) {
                const int kl = kk * 4 + laneHi * 2;
                v2f bf;
                bf.x = lz[(kl    ) * DD + c0 + laneRow];
                bf.y = lz[(kl + 1) * DD + c0 + laneRow];
                acc = wmma_f32(af[kk], bf, acc);
            }
            az[ct] = acc;
        }
        __syncthreads();
    }

    // ---- relu + gate partials ----
    float pg[8];
#pragma unroll
    for (int v = 0; v < 8; ++v) pg[v] = 0.0f;

#pragma unroll
    for (int ct = 0; ct < 8; ++ct) {
        const int c0 = ct * 16;
        const float gwz = gw[DD + c0 + laneRow];   // gate weight, az part
        const float gwx = gw[c0 + laneRow];        // gate weight, x part
#pragma unroll
        for (int v = 0; v < 8; ++v) {
            float a = fmaxf(az[ct][v], 0.0f);      // relu
            az[ct][v] = a;
            const int r = i0 + v + laneHi * 8;
            const float xv = x[zBase + (size_t)r * DD + c0 + laneRow];
            pg[v] += a * gwz + xv * gwx;
        }
    }

    // reduce gate partials over the 16 lanes that share a row group
#pragma unroll
    for (int v = 0; v < 8; ++v) {
        float s = pg[v];
        s += __shfl_xor(s, 1, 16);
        s += __shfl_xor(s, 2, 16);
        s += __shfl_xor(s, 4, 16);
        s += __shfl_xor(s, 8, 16);
        pg[v] = s;
    }
    const float gbias = gb[0];
    float coeff[8];
#pragma unroll
    for (int v = 0; v < 8; ++v)
        coeff[v] = 1.0f / (1.0f + __expf(-(pg[v] + gbias)));

#pragma unroll
    for (int ct = 0; ct < 8; ++ct) {
        const int c0 = ct * 16;
#pragma unroll
        for (int v = 0; v < 8; ++v) {
            const int r = i0 + v + laneHi * 8;
            const size_t idx = zBase + (size_t)r * DD + c0 + laneRow;
            zout[idx] = coeff[v] * x[idx] + (1.0f - coeff[v]) * az[ct][v];
        }
    }
}

// ---------------------------------------------------------------------------
extern "C" void kernel_launch(void* const* d_in, const int* in_sizes, int n_in,
                              void* d_out, int out_size, void* d_ws, size_t ws_size,
                              hipStream_t stream) {
    const float* x    = (const float*)d_in[0];   // [B,N,D]
    const float* adj  = (const float*)d_in[1];   // [B,N,N]
    const float* Ww   = (const float*)d_in[2];   // [D,D]
    const float* Wb   = (const float*)d_in[3];   // [D]
    const float* Amat = (const float*)d_in[4];   // [D,D]
    const float* gw   = (const float*)d_in[5];   // [1,2D]
    const float* gb   = (const float*)d_in[6];   // [1]
    float* out = (float*)d_out;                  // [B,N,D]

    float* ws = (float*)d_ws;
    float* h   = ws;                              //  2M floats
    float* hA  = h   + (size_t)BB * NN * DD;      //  2M floats
    float* att = hA  + (size_t)BB * NN * DD;      // 16M floats
    float* za  = att + (size_t)BB * NN * NN;      //  2M floats
    float* zb  = za  + (size_t)BB * NN * DD;      //  2M floats

    const int tiles = BB * (NN / 16);            // 1024

    gat_proj_kernel<<<tiles, 32, 0, stream>>>(x, Ww, Wb, Amat, h, hA);
    gat_scores_kernel<<<BB * 64 * 64, 32, 0, stream>>>(h, hA, adj, att);
    gat_softmax_kernel<<<BB * (NN / 256), 256, 0, stream>>>(adj, att);

    const int hopBlocks = BB * (NN / 128);       // 128 blocks of 8 waves
    // 4 hops, ping-pong: h -> za -> zb -> za -> out
    gat_hop_kernel<<<hopBlocks, 256, 0, stream>>>(att, x, h,  gw, gb, za);
    gat_hop_kernel<<<hopBlocks, 256, 0, stream>>>(att, x, za, gw, gb, zb);
    gat_hop_kernel<<<hopBlocks, 256, 0, stream>>>(att, x, zb, gw, gb, za);
    gat_hop_kernel<<<hopBlocks, 256, 0, stream>>>(att, x, za, gw, gb, out);
}